// EdgeSAGELayer_86474871537828
// MI455X (gfx1250) — compile-verified
//
#include <hip/hip_runtime.h>
#include <math.h>

#define N_NODES 100000
#define N_EDGES 625000
#define D 128

typedef __attribute__((ext_vector_type(2))) float v2f;
typedef __attribute__((ext_vector_type(8))) float v8f;

// ---------------------------------------------------------------------------
// Phase 0: zero the sum accumulator (aliased onto d_out) and the count array.
// ---------------------------------------------------------------------------
__global__ void zero_kernel(float* __restrict__ sums, float* __restrict__ cnt) {
    int idx = blockIdx.x * blockDim.x + threadIdx.x;
    const int total = N_NODES * D;
    for (int i = idx; i < total; i += gridDim.x * blockDim.x) sums[i] = 0.0f;
    if (idx < N_NODES) cnt[idx] = 0.0f;
}

// ---------------------------------------------------------------------------
// Phase 1: scatter-add. One wave32 per edge: coalesced float4 read of the
// 128-float edge feature row (512B per wave), 4 native f32 atomics per lane.
// ---------------------------------------------------------------------------
__global__ void __launch_bounds__(256) scatter_kernel(
        const float* __restrict__ edge_attr, const int* __restrict__ eidx,
        float* __restrict__ sums, float* __restrict__ cnt) {
    const int lane = threadIdx.x & 31;
    const int edge = (blockIdx.x * blockDim.x + threadIdx.x) >> 5;
    if (edge >= N_EDGES) return;

    const int t = eidx[edge];  // edge_index[0][e], wave-uniform
    const float4 v = ((const float4*)(edge_attr + (size_t)edge * D))[lane];
    float* dst = sums + (size_t)t * D + lane * 4;
    unsafeAtomicAdd(dst + 0, v.x);
    unsafeAtomicAdd(dst + 1, v.y);
    unsafeAtomicAdd(dst + 2, v.z);
    unsafeAtomicAdd(dst + 3, v.w);
    if (lane == 0) unsafeAtomicAdd(cnt + t, 1.0f);
}

// ---------------------------------------------------------------------------
// Phase 2: h = 0.5*(node + sums/max(cnt,1)); out = sigmoid(h @ W^T + b)
// using V_WMMA_F32_16X16X4_F32. Block = 256 thr (8 waves), 16 node rows per
// block; wave w computes output columns [16w, 16w+16). K is processed in two
// halves of 64 so W staging (128x68 words) + h tile (16x132 words) stay at
// ~43KB of LDS. Row strides 132/68 words => bank stride 4 => conflict-free
// b64 fragment reads.
// ---------------------------------------------------------------------------
__global__ void __launch_bounds__(256) gemm_kernel(
        const float* __restrict__ node_attr, const float* __restrict__ sums,
        const float* __restrict__ cnt, const float* __restrict__ Wm,
        const float* __restrict__ bias, float* __restrict__ out) {
    __shared__ float hlds[16 * 132];
    __shared__ float wlds[128 * 68];

    const int t   = threadIdx.x;
    const int r0  = blockIdx.x * 16;          // first node row of this block
    const int lane = t & 31;
    const int wv   = t >> 5;                  // 0..7 -> column tile
    const int am   = lane & 15;               // A row (M)
    const int koff = (lane >> 4) * 2;         // K sub-offset for A and B frags
    const int ocol = wv * 16 + (lane & 15);   // output column / W row

    // Stage h tile: h = node*0.5 + sums*(0.5/max(cnt,1))
    for (int i = t; i < 512; i += 256) {
        const int row = i >> 5, c4 = i & 31;
        const float4 s = ((const float4*)(sums      + (size_t)(r0 + row) * D))[c4];
        const float4 n = ((const float4*)(node_attr + (size_t)(r0 + row) * D))[c4];
        const float inv = 0.5f / fmaxf(cnt[r0 + row], 1.0f);
        float4 h;
        h.x = n.x * 0.5f + s.x * inv;
        h.y = n.y * 0.5f + s.y * inv;
        h.z = n.z * 0.5f + s.z * inv;
        h.w = n.w * 0.5f + s.w * inv;
        ((float4*)&hlds[row * 132])[c4] = h;   // byte addr row*528 + c4*16, 16B aligned
    }

    v8f acc = {0.f, 0.f, 0.f, 0.f, 0.f, 0.f, 0.f, 0.f};

    for (int kh = 0; kh < 2; ++kh) {
        __syncthreads();  // h ready (kh==0) / previous half consumed (kh==1)
        // Stage K-half of W: rows 0..127, cols kh*64..kh*64+63
        for (int i = t; i < 2048; i += 256) {
            const int row = i >> 4, c4 = i & 15;
            const float4 w4 = ((const float4*)(Wm + (size_t)row * D + kh * 64))[c4];
            ((float4*)&wlds[row * 68])[c4] = w4;  // byte addr row*272 + c4*16
        }
        __syncthreads();

        const float* ha = &hlds[am * 132 + kh * 64 + koff];
        const float* wb = &wlds[ocol * 68 + koff];
#pragma unroll
        for (int k0 = 0; k0 < 64; k0 += 4) {
            const v2f a = *(const v2f*)(ha + k0);   // A[M=am][k0+koff .. +1]
            const v2f b = *(const v2f*)(wb + k0);   // B[k0+koff..+1][N=ocol] = W[ocol][..]
            acc = __builtin_amdgcn_wmma_f32_16x16x4_f32(
                false, a, false, b, (short)0, acc, false, false);
        }
    }

    // Epilogue: C/D layout -> lane holds N = lane%16, rows M = j + 8*(lane/16)
    const float bb = bias[ocol];
#pragma unroll
    for (int j = 0; j < 8; ++j) {
        const int m = j + 8 * (lane >> 4);
        const float x = acc[j] + bb;
        out[(size_t)(r0 + m) * D + ocol] = 1.0f / (1.0f + __expf(-x));
    }
}

// ---------------------------------------------------------------------------
extern "C" void kernel_launch(void* const* d_in, const int* in_sizes, int n_in,
                              void* d_out, int out_size, void* d_ws, size_t ws_size,
                              hipStream_t stream) {
    const float* node_attr = (const float*)d_in[0];
    const float* edge_attr = (const float*)d_in[1];
    const int*   eidx      = (const int*)d_in[2];   // row 0 = targets
    const float* Wm        = (const float*)d_in[3]; // [128,128] row-major
    const float* bias      = (const float*)d_in[4];

    float* out = (float*)d_out;      // doubles as scatter-sum accumulator
    float* cnt = (float*)d_ws;       // N_NODES floats of workspace

    zero_kernel<<<(N_NODES * D + 255) / 256, 256, 0, stream>>>(out, cnt);
    scatter_kernel<<<(N_EDGES * 32 + 255) / 256, 256, 0, stream>>>(edge_attr, eidx, out, cnt);
    gemm_kernel<<<N_NODES / 16, 256, 0, stream>>>(node_attr, out, cnt, Wm, bias, out);
}